// PairLoss_20272245637747
// MI455X (gfx1250) — compile-verified
//
#include <hip/hip_runtime.h>
#include <math.h>

// D = A(16x4 f32) * B(4x16 f32) + C(16x16 f32), striped across the wave32.
// A fragment = 2 VGPRs/lane (64 values per wave), C/D = 8 VGPRs/lane.
typedef __attribute__((ext_vector_type(2))) float v2f;
typedef __attribute__((ext_vector_type(4))) float v4f;
typedef __attribute__((ext_vector_type(8))) float v8f;

#define BLOCK 256
#define WAVES (BLOCK / 32)

__global__ __launch_bounds__(BLOCK) void pairloss_partials(
    const float* __restrict__ pred, const float* __restrict__ gt,
    float* __restrict__ wsA, float* __restrict__ wsB,
    unsigned* __restrict__ wsC, int n) {
  const int tid = blockIdx.x * blockDim.x + threadIdx.x;
  const int nthreads = gridDim.x * blockDim.x;
  const int nvec = n >> 2;               // float4 elements
  const int full = nvec / nthreads;      // iterations where every lane is valid

  v8f accA = {};  // sum exp(pred) over negatives
  v8f accB = {};  // sum exp(-pred) over positives
  unsigned cnt = 0;
  const v2f bones = {1.0f, 1.0f};  // B = all-ones -> D accumulates row sums

  const v4f* __restrict__ pred4 = (const v4f*)pred;
  const v4f* __restrict__ gt4 = (const v4f*)gt;

  // Steady state: unconditional non-temporal b128 loads (single-pass stream,
  // 256 MB > 192 MB L2 -> NT avoids cache thrash), no EXEC manipulation.
  for (int k = 0; k < full; ++k) {
    const int i = tid + k * nthreads;
    const v4f p = __builtin_nontemporal_load(&pred4[i]);
    const v4f g = __builtin_nontemporal_load(&gt4[i]);
    float pv[4] = {p.x, p.y, p.z, p.w};
    float gv[4] = {g.x, g.y, g.z, g.w};
    float en[4], ep[4];
#pragma unroll
    for (int e = 0; e < 4; ++e) {
      const bool neg = gv[e] < 0.5f;
      // One transcendental per element, branch-free: select the argument,
      // not the result (v_cndmask + v_exp_f32; EXEC stays all-ones).
      const float t = expf(neg ? pv[e] : -pv[e]);
      en[e] = neg ? t : 0.0f;
      ep[e] = neg ? 0.0f : t;
      cnt += neg ? 1u : 0u;
    }
    // Route the reduction adds through the matrix pipe at full f32 precision.
    v2f a0 = {en[0], en[1]}, a1 = {en[2], en[3]};
    v2f b0 = {ep[0], ep[1]}, b1 = {ep[2], ep[3]};
    accA = __builtin_amdgcn_wmma_f32_16x16x4_f32(false, a0, false, bones,
                                                 (short)0, accA, false, false);
    accA = __builtin_amdgcn_wmma_f32_16x16x4_f32(false, a1, false, bones,
                                                 (short)0, accA, false, false);
    accB = __builtin_amdgcn_wmma_f32_16x16x4_f32(false, b0, false, bones,
                                                 (short)0, accB, false, false);
    accB = __builtin_amdgcn_wmma_f32_16x16x4_f32(false, b1, false, bones,
                                                 (short)0, accB, false, false);
  }

  // Single guarded epilogue iteration (uniform trip: all threads reach the
  // WMMAs with EXEC all-ones; only the loads are predicated).
  {
    const int i = tid + full * nthreads;
    const bool valid = (i < nvec);
    v4f p = {0.f, 0.f, 0.f, 0.f};
    v4f g = {1.f, 1.f, 1.f, 1.f};
    if (valid) {
      p = pred4[i];
      g = gt4[i];
    }
    float pv[4] = {p.x, p.y, p.z, p.w};
    float gv[4] = {g.x, g.y, g.z, g.w};
    float en[4], ep[4];
#pragma unroll
    for (int e = 0; e < 4; ++e) {
      const bool neg = gv[e] < 0.5f;
      const bool pos = gv[e] > 0.5f;
      const float t = expf(neg ? pv[e] : -pv[e]);
      en[e] = (valid && neg) ? t : 0.0f;
      ep[e] = (valid && pos) ? t : 0.0f;
      cnt += (valid && neg) ? 1u : 0u;
    }
    v2f a0 = {en[0], en[1]}, a1 = {en[2], en[3]};
    v2f b0 = {ep[0], ep[1]}, b1 = {ep[2], ep[3]};
    accA = __builtin_amdgcn_wmma_f32_16x16x4_f32(false, a0, false, bones,
                                                 (short)0, accA, false, false);
    accA = __builtin_amdgcn_wmma_f32_16x16x4_f32(false, a1, false, bones,
                                                 (short)0, accA, false, false);
    accB = __builtin_amdgcn_wmma_f32_16x16x4_f32(false, b0, false, bones,
                                                 (short)0, accB, false, false);
    accB = __builtin_amdgcn_wmma_f32_16x16x4_f32(false, b1, false, bones,
                                                 (short)0, accB, false, false);
  }

  // Every D column holds the full row-sums; lane l has M=0..7 of column l,
  // lane l+16 has M=8..15. One xor-16 shuffle gives the wave grand total.
  float sA = accA[0] + accA[1] + accA[2] + accA[3] + accA[4] + accA[5] +
             accA[6] + accA[7];
  float sB = accB[0] + accB[1] + accB[2] + accB[3] + accB[4] + accB[5] +
             accB[6] + accB[7];
  sA += __shfl_xor(sA, 16, 32);
  sB += __shfl_xor(sB, 16, 32);
#pragma unroll
  for (int off = 16; off > 0; off >>= 1) cnt += __shfl_xor(cnt, off, 32);

  __shared__ float lA[WAVES];
  __shared__ float lB[WAVES];
  __shared__ unsigned lC[WAVES];
  const int wave = threadIdx.x >> 5;
  const int lane = threadIdx.x & 31;
  if (lane == 0) { lA[wave] = sA; lB[wave] = sB; lC[wave] = cnt; }
  __syncthreads();
  if (threadIdx.x == 0) {
    float tA = 0.f, tB = 0.f;
    unsigned tC = 0;
    for (int w = 0; w < WAVES; ++w) { tA += lA[w]; tB += lB[w]; tC += lC[w]; }
    wsA[blockIdx.x] = tA;
    wsB[blockIdx.x] = tB;
    wsC[blockIdx.x] = tC;
  }
}

__global__ __launch_bounds__(BLOCK) void pairloss_finalize(
    const float* __restrict__ wsA, const float* __restrict__ wsB,
    const unsigned* __restrict__ wsC, int nparts,
    const float* __restrict__ pred, const float* __restrict__ gt, int n,
    float* __restrict__ out) {
  __shared__ float lA[BLOCK];
  __shared__ float lB[BLOCK];
  __shared__ unsigned lC[BLOCK];
  const int t = threadIdx.x;
  float a = 0.f, b = 0.f;
  unsigned c = 0;
  for (int i = t; i < nparts; i += BLOCK) { a += wsA[i]; b += wsB[i]; c += wsC[i]; }
  lA[t] = a; lB[t] = b; lC[t] = c;
  __syncthreads();
  if (t == 0) {
    float tA = 0.f, tB = 0.f;
    unsigned tC = 0;
    for (int i = 0; i < BLOCK; ++i) { tA += lA[i]; tB += lB[i]; tC += lC[i]; }
    // Scalar tail for n not divisible by 4 (N=2^25 has none; kept for safety).
    for (int i = (n & ~3); i < n; ++i) {
      const float g = gt[i], p = pred[i];
      if (g < 0.5f) { tA += expf(p); tC += 1u; }
      else if (g > 0.5f) { tB += expf(-p); }
    }
    out[0] = (tA * tB - (float)tC) * 0.5f;
  }
}

extern "C" void kernel_launch(void* const* d_in, const int* in_sizes, int n_in,
                              void* d_out, int out_size, void* d_ws,
                              size_t ws_size, hipStream_t stream) {
  const float* pred = (const float*)d_in[0];
  const float* gt = (const float*)d_in[1];
  float* out = (float*)d_out;
  const int n = in_sizes[0];

  int grid = 1024;  // 1024 blocks x 8 waves: plenty to saturate 23.3 TB/s HBM
  const size_t per_block = 2 * sizeof(float) + sizeof(unsigned);
  if (ws_size < (size_t)grid * per_block) {
    grid = (int)(ws_size / per_block);
    if (grid < 1) grid = 1;
  }
  float* wsA = (float*)d_ws;
  float* wsB = wsA + grid;
  unsigned* wsC = (unsigned*)(wsB + grid);

  pairloss_partials<<<grid, BLOCK, 0, stream>>>(pred, gt, wsA, wsB, wsC, n);
  pairloss_finalize<<<1, BLOCK, 0, stream>>>(wsA, wsB, wsC, grid, pred, gt, n,
                                             out);
}